// InvariantsTransIso_74131135529061
// MI455X (gfx1250) — compile-verified
//
#include <hip/hip_runtime.h>

// ---------------------------------------------------------------------------
// InvariantsTransIso: per 3x3 matrix F compute
//   C   = F^T F                      (symmetric)
//   I1  = tr(C)
//   J   = det(F)
//   I4  = sum_ij C_ij  G_ji
//   I5  = sum_ij cofC_ij G_ji        (cofC symmetric since C symmetric)
// out[n] = {I1, J, -J, I4, I5}
//
// Memory-bound streaming kernel (56 B and ~70 FLOP per matrix -> ~1.2 F/B,
// HBM-limited at 23.3 TB/s; ~224 MB total => ~9.6 us floor). CDNA5 path:
// async global<->LDS b128 copies (ASYNCcnt, inline asm) with NT temporal
// hints (stream > 192 MB L2) for perfectly coalesced 128-bit streaming of
// the awkward 36 B-in / 20 B-out per-item records. WMMA is intentionally
// not used: batched 3x3 products have no efficient 16x16xK mapping and the
// kernel is bandwidth-bound regardless.
// ---------------------------------------------------------------------------

typedef float v4f __attribute__((ext_vector_type(4)));

#define TPB   256            // threads per block (8 waves, wave32)
#define MPB   256            // matrices per block
// input stage: 256*9 floats = 576 x b128 (9216 B, block base 16B-aligned)
// output stage: 256*5 floats = 320 x b128 (5120 B, block base 16B-aligned)

// ---- CDNA5 async global<->LDS copies (ISA 15.18 VGLOBAL, ASYNCcnt) --------
// LDS byte address = low 32 bits of the generic shared-pointer address
// (aperture lives in addr[63:32]).

__device__ __forceinline__ void async_load_b128(const void* gptr, void* lptr) {
    unsigned int       lds = (unsigned int)(uintptr_t)lptr;
    unsigned long long g   = (unsigned long long)(uintptr_t)gptr;
    asm volatile("global_load_async_to_lds_b128 %0, %1, off th:TH_LOAD_NT"
                 :: "v"(lds), "v"(g)
                 : "memory");
}

__device__ __forceinline__ void async_store_b128(void* gptr, const void* lptr) {
    unsigned int       lds = (unsigned int)(uintptr_t)lptr;
    unsigned long long g   = (unsigned long long)(uintptr_t)gptr;
    asm volatile("global_store_async_from_lds_b128 %0, %1, off th:TH_STORE_NT"
                 :: "v"(g), "v"(lds)
                 : "memory");
}

__device__ __forceinline__ void wait_async0() {
    asm volatile("s_wait_asynccnt 0" ::: "memory");
}

__global__ void __launch_bounds__(TPB)
invariants_transiso_kernel(const float* __restrict__ F,
                           const float* __restrict__ G,
                           float* __restrict__ out,
                           int n)
{
    __shared__ float sIn[MPB * 9];   // 9216 B staged input
    __shared__ float sOut[MPB * 5];  // 5120 B staged output

    const int tid = threadIdx.x;
    const long long mbase = (long long)blockIdx.x * MPB;
    const int nFull = n / MPB;

    // G is uniform across the grid -> scalar (s_load / s_add_f32) math.
    const float g00 = G[0], g11 = G[4], g22 = G[8];
    const float gs01 = G[1] + G[3];   // weight for C01 (+C10)
    const float gs02 = G[2] + G[6];   // weight for C02 (+C20)
    const float gs12 = G[5] + G[7];   // weight for C12 (+C21)

    if ((int)blockIdx.x < nFull) {
        // ---- stage input: 576 x 16B = 2 full rounds + 64, async into LDS --
        const v4f* __restrict__ g4 = (const v4f*)(F + mbase * 9);
        v4f* s4 = (v4f*)sIn;
        async_load_b128(g4 + tid,       s4 + tid);
        async_load_b128(g4 + 256 + tid, s4 + 256 + tid);
        if (tid < 64) {
            async_load_b128(g4 + 512 + tid, s4 + 512 + tid);
        }
        wait_async0();
        __syncthreads();

        // ---- per-thread 3x3 invariants (LDS stride 9: bank-conflict free) --
        const float* m = &sIn[tid * 9];
        const float f00 = m[0], f01 = m[1], f02 = m[2];
        const float f10 = m[3], f11 = m[4], f12 = m[5];
        const float f20 = m[6], f21 = m[7], f22 = m[8];

        // C = F^T F (columns of F dotted)
        const float C00 = f00*f00 + f10*f10 + f20*f20;
        const float C11 = f01*f01 + f11*f11 + f21*f21;
        const float C22 = f02*f02 + f12*f12 + f22*f22;
        const float C01 = f00*f01 + f10*f11 + f20*f21;
        const float C02 = f00*f02 + f10*f12 + f20*f22;
        const float C12 = f01*f02 + f11*f12 + f21*f22;

        const float I1 = C00 + C11 + C22;
        const float J  = f00*(f11*f22 - f12*f21)
                       - f01*(f10*f22 - f12*f20)
                       + f02*(f10*f21 - f11*f20);

        const float I4 = C00*g00 + C11*g11 + C22*g22
                       + C01*gs01 + C02*gs02 + C12*gs12;

        // cofactor of symmetric C (also symmetric)
        const float K00 = C11*C22 - C12*C12;
        const float K11 = C00*C22 - C02*C02;
        const float K22 = C00*C11 - C01*C01;
        const float K01 = C12*C02 - C01*C22;
        const float K02 = C01*C12 - C11*C02;
        const float K12 = C02*C01 - C12*C00;

        const float I5 = K00*g00 + K11*g11 + K22*g22
                       + K01*gs01 + K02*gs02 + K12*gs12;

        // ---- stage output (LDS stride 5: bank-conflict free) ----
        float* o = &sOut[tid * 5];
        o[0] = I1; o[1] = J; o[2] = -J; o[3] = I4; o[4] = I5;
        __syncthreads();

        // ---- stream out: 320 x 16B = 1 full round + 64, async from LDS ----
        v4f* go4 = (v4f*)(out + mbase * 5);
        v4f* so4 = (v4f*)sOut;
        async_store_b128(go4 + tid, so4 + tid);
        if (tid < 64) {
            async_store_b128(go4 + 256 + tid, so4 + 256 + tid);
        }
        // S_ENDPGM performs an implicit wait-idle, but wait explicitly so the
        // LDS buffers are provably quiescent before workgroup teardown.
        wait_async0();
    } else {
        // ---- tail block (never taken for N = 4,000,000): direct path ----
        const long long t = mbase + tid;
        if (t < (long long)n) {
            const float* m = F + t * 9;
            const float f00 = m[0], f01 = m[1], f02 = m[2];
            const float f10 = m[3], f11 = m[4], f12 = m[5];
            const float f20 = m[6], f21 = m[7], f22 = m[8];

            const float C00 = f00*f00 + f10*f10 + f20*f20;
            const float C11 = f01*f01 + f11*f11 + f21*f21;
            const float C22 = f02*f02 + f12*f12 + f22*f22;
            const float C01 = f00*f01 + f10*f11 + f20*f21;
            const float C02 = f00*f02 + f10*f12 + f20*f22;
            const float C12 = f01*f02 + f11*f12 + f21*f22;

            const float I1 = C00 + C11 + C22;
            const float J  = f00*(f11*f22 - f12*f21)
                           - f01*(f10*f22 - f12*f20)
                           + f02*(f10*f21 - f11*f20);
            const float I4 = C00*g00 + C11*g11 + C22*g22
                           + C01*gs01 + C02*gs02 + C12*gs12;
            const float K00 = C11*C22 - C12*C12;
            const float K11 = C00*C22 - C02*C02;
            const float K22 = C00*C11 - C01*C01;
            const float K01 = C12*C02 - C01*C22;
            const float K02 = C01*C12 - C11*C02;
            const float K12 = C02*C01 - C12*C00;
            const float I5 = K00*g00 + K11*g11 + K22*g22
                           + K01*gs01 + K02*gs02 + K12*gs12;

            float* o = out + t * 5;
            o[0] = I1; o[1] = J; o[2] = -J; o[3] = I4; o[4] = I5;
        }
    }
}

extern "C" void kernel_launch(void* const* d_in, const int* in_sizes, int n_in,
                              void* d_out, int out_size, void* d_ws, size_t ws_size,
                              hipStream_t stream) {
    (void)n_in; (void)out_size; (void)d_ws; (void)ws_size;
    const float* F = (const float*)d_in[0];
    const float* G = (const float*)d_in[1];
    float* out = (float*)d_out;
    const int n = in_sizes[0] / 9;          // number of 3x3 matrices
    const int grid = (n + MPB - 1) / MPB;   // 15625 full blocks for N = 4e6
    invariants_transiso_kernel<<<grid, TPB, 0, stream>>>(F, G, out, n);
}